// SimpleVectorQuantizer_21483426415069
// MI455X (gfx1250) — compile-verified
//
#include <hip/hip_runtime.h>

typedef __attribute__((ext_vector_type(2))) float v2f;
typedef __attribute__((ext_vector_type(8))) float v8f;

#define NROWS   16384      // 16 * 1024 flattened z rows
#define NCODES  8192
#define DIM     256
#define BM      128        // rows per workgroup tile (resident)
#define BN      64         // codes per chunk (double buffered)
#define NCHUNK  (NCODES / BN)          // 128 (even: loop unrolled by 2)
#define LDA     260        // padded LDS row stride (floats): row stride = 4 banks
#define LDB     260
#define BUFF    (BN * LDB)             // floats per B buffer
#define BIGF    3.4e38f

// ---- CDNA5 async global->LDS (ASYNCcnt) helpers --------------------------
__device__ __forceinline__ void async_lds_b128(unsigned ldsoff, unsigned goff,
                                               unsigned long long base) {
    // GVS form: copies 16 bytes from the global address base + goff to the LDS address in ldsoff, tracked by ASYNCcnt
    asm volatile("global_load_async_to_lds_b128 %0, %1, %2"
                 :: "v"(ldsoff), "v"(goff), "s"(base) : "memory");
}
__device__ __forceinline__ void wait_async0() {
    asm volatile("s_wait_asynccnt 0x0" ::: "memory");
}

// ---------------------------------------------------------------------------
// Kernel 1: e2[k] = ||embedding[k]||^2   (one row per wave)
// ---------------------------------------------------------------------------
__global__ __launch_bounds__(256)
void vq_e2_kernel(const float* __restrict__ emb, float* __restrict__ e2) {
    const int wave = threadIdx.x >> 5;
    const int lane = threadIdx.x & 31;
    const int row  = blockIdx.x * 8 + wave;
    const float4* r4 = (const float4*)(emb + (size_t)row * DIM);
    float4 a = r4[lane];
    float4 b = r4[lane + 32];
    float s = a.x*a.x + a.y*a.y + a.z*a.z + a.w*a.w
            + b.x*b.x + b.y*b.y + b.z*b.z + b.w*b.w;
    #pragma unroll
    for (int off = 16; off > 0; off >>= 1) s += __shfl_xor(s, off, 32);
    if (lane == 0) e2[row] = s;
}

// ---------------------------------------------------------------------------
// Kernel 2: S = Z * E^T via V_WMMA_F32_16X16X4_F32 with double-buffered
//           async global->LDS staging of the codebook; fused per-row argmin
//           of (e2[k] - 2*S[n,k]).  ~267KB LDS per workgroup (320KB WGP).
//           Chunk loop unrolled by 2 so both LDS buffers have compile-time
//           bases -> pure ds_load_2addr_b64 fragment reads (no flat loads).
// ---------------------------------------------------------------------------

// issue async DMA of chunk NEXT into buffer PAR (0/1)
#define VQ_ISSUE(NEXT, PAR)                                                   \
    {                                                                         \
        unsigned l = ldsBase0 + (unsigned)((PAR) * BUFF * 4);                 \
        unsigned g = (unsigned)((((NEXT) * BN + r0) * DIM + c4 * 4) * 4);     \
        _Pragma("unroll")                                                     \
        for (int j = 0; j < 16; ++j) {                                        \
            async_lds_b128(l, g, embBase);                                    \
            l += 4 * LDB * 4;                                                 \
            g += 4 * DIM * 4;                                                 \
        }                                                                     \
        if (tid < BN) e2s[(PAR) * BN + tid] = e2[(NEXT) * BN + tid];          \
    }

// WMMA burst over chunk CHUNK from fragment base BF / e2 base EZ, then fold
#define VQ_STEP(CHUNK, BF, EZ)                                                \
    {                                                                         \
        v8f acc[2][2] = {};                                                   \
        _Pragma("unroll 8")                                                   \
        for (int k = 0; k < DIM; k += 4) {                                    \
            v2f a0 = *(const v2f*)(Afrag + k);                                \
            v2f a1 = *(const v2f*)(Afrag + 16 * LDA + k);                     \
            v2f b0 = *(const v2f*)((BF) + k);                                 \
            v2f b1 = *(const v2f*)((BF) + 16 * LDB + k);                      \
            acc[0][0] = __builtin_amdgcn_wmma_f32_16x16x4_f32(false, a0, false, b0, (short)0, acc[0][0], false, false); \
            acc[0][1] = __builtin_amdgcn_wmma_f32_16x16x4_f32(false, a0, false, b1, (short)0, acc[0][1], false, false); \
            acc[1][0] = __builtin_amdgcn_wmma_f32_16x16x4_f32(false, a1, false, b0, (short)0, acc[1][0], false, false); \
            acc[1][1] = __builtin_amdgcn_wmma_f32_16x16x4_f32(false, a1, false, b1, (short)0, acc[1][1], false, false); \
        }                                                                     \
        const int   colbase = (CHUNK) * BN + wn * 32 + l15;                   \
        const float eq0 = (EZ)[0];                                            \
        const float eq1 = (EZ)[16];                                           \
        _Pragma("unroll")                                                     \
        for (int tm = 0; tm < 2; ++tm) {                                      \
            _Pragma("unroll")                                                 \
            for (int r = 0; r < 8; ++r) {                                     \
                float d0 = __builtin_fmaf(-2.0f, acc[tm][0][r], eq0);         \
                bool  t0 = d0 < bestv[tm][r];                                 \
                bestv[tm][r] = t0 ? d0 : bestv[tm][r];                        \
                besti[tm][r] = t0 ? colbase : besti[tm][r];                   \
                float d1 = __builtin_fmaf(-2.0f, acc[tm][1][r], eq1);         \
                bool  t1 = d1 < bestv[tm][r];                                 \
                bestv[tm][r] = t1 ? d1 : bestv[tm][r];                        \
                besti[tm][r] = t1 ? (colbase + 16) : besti[tm][r];            \
            }                                                                 \
        }                                                                     \
    }

__global__ __launch_bounds__(256)
void vq_argmin_kernel(const float* __restrict__ z, const float* __restrict__ emb,
                      const float* __restrict__ e2, int* __restrict__ idx_out,
                      float* __restrict__ idx_out_f) {
    extern __shared__ float smem[];
    float* Bs  = smem;                  // 2 x [BN][LDB] double-buffered E chunk
    float* e2s = Bs + 2 * BUFF;         // 2 x [BN]
    float* As  = e2s + 2 * BN;          // [BM][LDA]   resident Z tile

    const int tid  = threadIdx.x;
    const int lane = tid & 31;
    const int wave = tid >> 5;
    const int wm   = wave & 3;          // 4 row groups  -> 32 rows each
    const int wn   = wave >> 2;         // 2 col groups  -> 32 cols each
    const int m0   = blockIdx.x * BM;
    const int hi   = lane >> 4;         // 0: lanes 0-15, 1: lanes 16-31
    const int l15  = lane & 15;

    // Per-thread async slice: column block c4 fixed, rows r0, r0+4, ..., r0+60
    const int r0 = tid >> 6;
    const int c4 = tid & 63;
    // LDS aperture keeps the byte offset in addr[31:0]
    const unsigned BsOff    = (unsigned)(size_t)(void*)Bs;
    const unsigned ldsBase0 = BsOff + (unsigned)((r0 * LDB + c4 * 4) * 4);
    const unsigned long long embBase = (unsigned long long)(const void*)emb;

    // ---- prologue: kick off chunk 0 DMA, then stage A while it flies ----
    VQ_ISSUE(0, 0);
    {
        const float4* zg = (const float4*)(z + (size_t)m0 * DIM);
        #pragma unroll 4
        for (int i = 0; i < 32; ++i) {  // 8192 float4 / 256 threads
            int linear = tid + 256 * i;
            int row = linear >> 6;
            int cc  = linear & 63;
            float4 v = zg[row * 64 + cc];
            float* dst = As + row * LDA + cc * 4;
            dst[0] = v.x; dst[1] = v.y; dst[2] = v.z; dst[3] = v.w;
        }
    }

    float bestv[2][8];
    int   besti[2][8];
    #pragma unroll
    for (int tm = 0; tm < 2; ++tm)
        #pragma unroll
        for (int r = 0; r < 8; ++r) { bestv[tm][r] = BIGF; besti[tm][r] = 0; }

    // Fragment bases (ISA 16x4 f32 A/B layout: lane holds K = 2*hi + j).
    // Compile-time-constant buffer offsets keep these in LDS address space.
    const float* Afrag = As + (wm * 32 + l15) * LDA + 2 * hi;
    const float* Bf0   = Bs + (wn * 32 + l15) * LDB + 2 * hi;
    const float* Bf1   = Bf0 + BUFF;
    const float* ez0   = e2s + wn * 32 + l15;
    const float* ez1   = ez0 + BN;

    for (int i = 0; i < NCHUNK; i += 2) {
        // ---- even chunk: compute buf0, DMA next into buf1 ----
        wait_async0();                  // own chunk-i loads landed
        __syncthreads();                // everyone's landed; buffers swappable
        VQ_ISSUE(i + 1, 1);             // i+1 <= 127 always
        VQ_STEP(i, Bf0, ez0);

        // ---- odd chunk: compute buf1, DMA next into buf0 ----
        wait_async0();
        __syncthreads();
        if (i + 2 < NCHUNK) VQ_ISSUE(i + 2, 0);
        VQ_STEP(i + 1, Bf1, ez1);
    }

    // ---- Branchless reduce across the 16 lanes holding the same rows ----
    #pragma unroll
    for (int tm = 0; tm < 2; ++tm)
        #pragma unroll
        for (int r = 0; r < 8; ++r) {
            float v = bestv[tm][r];
            int   i = besti[tm][r];
            #pragma unroll
            for (int off = 1; off < 16; off <<= 1) {  // stays inside 16-lane halves
                float ov = __shfl_xor(v, off, 32);
                int   oi = __shfl_xor(i, off, 32);
                bool take = (ov < v) | ((ov == v) & (oi < i));
                v = take ? ov : v;
                i = take ? oi : i;
            }
            bestv[tm][r] = v; besti[tm][r] = i;
        }

    // ---- Combine the two wave-column groups via LDS (reuse Bs region) ----
    __syncthreads();
    float* rv = smem;                   // [2][BM]
    int*   ri = (int*)(smem + 2 * BM);  // [2][BM]
    if (l15 == 0) {
        #pragma unroll
        for (int tm = 0; tm < 2; ++tm)
            #pragma unroll
            for (int r = 0; r < 8; ++r) {
                int lrow = wm * 32 + tm * 16 + r + 8 * hi;
                rv[wn * BM + lrow] = bestv[tm][r];
                ri[wn * BM + lrow] = besti[tm][r];
            }
    }
    __syncthreads();
    if (tid < BM) {
        float v0 = rv[tid];      int i0 = ri[tid];
        float v1 = rv[BM + tid]; int i1 = ri[BM + tid];
        bool take = (v1 < v0) | ((v1 == v0) & (i1 < i0));
        int best = take ? i1 : i0;
        idx_out  [m0 + tid] = best;
        idx_out_f[m0 + tid] = (float)best;
    }
}

// ---------------------------------------------------------------------------
// Kernel 3: z_q gather (straight-through forward value == z_q) + loss partial
// ---------------------------------------------------------------------------
__global__ __launch_bounds__(256)
void vq_gather_kernel(const float* __restrict__ z, const float* __restrict__ emb,
                      const int* __restrict__ idx, float* __restrict__ zq,
                      float* __restrict__ acc) {
    const int wave = threadIdx.x >> 5;
    const int lane = threadIdx.x & 31;
    const int row  = blockIdx.x * 8 + wave;
    const int code = idx[row];
    const float4* e4 = (const float4*)(emb + (size_t)code * DIM);
    const float4* z4 = (const float4*)(z + (size_t)row * DIM);
    float4*       o4 = (float4*)(zq + (size_t)row * DIM);
    float s = 0.f;
    #pragma unroll
    for (int j = 0; j < 2; ++j) {
        int c = lane + 32 * j;
        float4 e  = e4[c];
        float4 zz = z4[c];
        o4[c] = e;
        float dx = e.x - zz.x, dy = e.y - zz.y, dz = e.z - zz.z, dw = e.w - zz.w;
        s += dx*dx + dy*dy + dz*dz + dw*dw;
    }
    #pragma unroll
    for (int off = 16; off > 0; off >>= 1) s += __shfl_xor(s, off, 32);
    if (lane == 0) atomicAdd(acc, s);
}

// ---------------------------------------------------------------------------
// Kernel 4: vq_loss = mean((zq - z)^2) * 0.25
// ---------------------------------------------------------------------------
__global__ void vq_finalize_kernel(const float* __restrict__ acc,
                                   float* __restrict__ out_loss) {
    *out_loss = (*acc) * (0.25f / (float)(NROWS * DIM));
}

// ---------------------------------------------------------------------------
extern "C" void kernel_launch(void* const* d_in, const int* in_sizes, int n_in,
                              void* d_out, int out_size, void* d_ws, size_t ws_size,
                              hipStream_t stream) {
    (void)in_sizes; (void)n_in; (void)out_size; (void)ws_size;
    const float* z   = (const float*)d_in[0];   // (16,1024,256) f32
    const float* emb = (const float*)d_in[1];   // (8192,256)   f32

    float* out  = (float*)d_out;
    float* zq   = out;                                 // 4,194,304 floats
    float* idxf = out + (size_t)NROWS * DIM;           // 16,384 floats
    float* loss = idxf + NROWS;                        // 1 float

    float* e2   = (float*)d_ws;                        // 8192 floats
    int*   idxi = (int*)(e2 + NCODES);                 // 16384 ints
    float* acc  = (float*)(idxi + NROWS);              // 1 float

    vq_e2_kernel<<<NCODES / 8, 256, 0, stream>>>(emb, e2);

    const size_t lds_bytes =
        (size_t)(2 * BUFF + 2 * BN + BM * LDA) * sizeof(float);  // ~267 KB
    hipFuncSetAttribute((const void*)vq_argmin_kernel,
                        hipFuncAttributeMaxDynamicSharedMemorySize, (int)lds_bytes);
    vq_argmin_kernel<<<NROWS / BM, 256, lds_bytes, stream>>>(z, emb, e2, idxi, idxf);

    hipMemsetAsync(acc, 0, sizeof(float), stream);
    vq_gather_kernel<<<NROWS / 8, 256, 0, stream>>>(z, emb, idxi, zq, acc);
    vq_finalize_kernel<<<1, 1, 0, stream>>>(acc, loss);
}